// GreenCatNLLLoss_80934363726218
// MI455X (gfx1250) — compile-verified
//
#include <hip/hip_runtime.h>
#include <stdint.h>

// ---------------------------------------------------------------------------
// GreenCatNLLLoss for MI455X (gfx1250, wave32)
// Streaming reduction: 216 MB in, 1 float out. HBM-bound (~9.5us floor).
// Data path: coalesced GLOBAL_LOAD_ASYNC_TO_LDS_B64 staging (ASYNCcnt),
// conflict-free DS_LOAD_B64 row scans, deterministic two-level reduction.
// ---------------------------------------------------------------------------

namespace {

constexpr int kC           = 54;   // columns (fixed by reference)
constexpr int kChunks      = 27;   // 8-byte chunks per row (54 floats)
constexpr int kRowsPerTile = 32;   // one wave32 per tile
constexpr int kWavesPerBlk = 8;    // 256 threads
constexpr int kLdsStride   = 58;   // floats; 58 = -6 mod 64 -> bank-conflict-free

// categories {6, 11, 21..32, 38, 39, 44..52} as a 64-bit mask
constexpr unsigned long long kKeptMask =
    (1ull << 6) | (1ull << 11) |
    (((1ull << 12) - 1) << 21) |   // 21..32
    (3ull << 38) |                 // 38, 39
    (((1ull << 9) - 1) << 44);     // 44..52

// pointee types matching the async-load builtin's signature:
// (AS1 v2i32*, AS3 v2i32*, imm offset, imm cpol)
typedef int v2i __attribute__((vector_size(8)));
typedef __attribute__((address_space(1))) v2i v2i_glob;
typedef __attribute__((address_space(3))) v2i v2i_lds;

__device__ __forceinline__ void async_load_b64_to_lds(const float* gp, float* lp) {
#if __has_builtin(__builtin_amdgcn_global_load_async_to_lds_b64)
  __builtin_amdgcn_global_load_async_to_lds_b64(
      (v2i_glob*)(void*)(gp),
      (v2i_lds*)(void*)(lp),
      /*offset=*/0, /*cpol=*/0);
#else
  unsigned loff = (unsigned)(unsigned long long)
      (__attribute__((address_space(3))) char*)(lp);
  asm volatile("global_load_async_to_lds_b64 %0, %1, off"
               :: "v"(loff), "v"((unsigned long long)gp)
               : "memory");
#endif
}

__device__ __forceinline__ void wait_async0() {
#if __has_builtin(__builtin_amdgcn_s_wait_asynccnt)
  __builtin_amdgcn_s_wait_asynccnt(0);
#else
  asm volatile("s_wait_asynccnt 0" ::: "memory");
#endif
}

__device__ __forceinline__ void wait_ds0() {
#if __has_builtin(__builtin_amdgcn_s_wait_dscnt)
  __builtin_amdgcn_s_wait_dscnt(0);
#else
  asm volatile("s_wait_dscnt 0" ::: "memory");
#endif
}

} // namespace

// ---------------------------------------------------------------------------
// Kernel 1: per-wave tile processing, per-block partial sum of log(gathered)
// ---------------------------------------------------------------------------
__global__ void __launch_bounds__(256)
green_nll_main(const float* __restrict__ out, const int* __restrict__ tgt,
               float* __restrict__ partials, unsigned n, unsigned tiles,
               float adj) {
  // 8 waves x 32 rows x 58 floats = 59392 B, + 1 KB reduction scratch
  __shared__ float tile[kWavesPerBlk][kRowsPerTile * kLdsStride];
  __shared__ float red[256];

  const unsigned wave = threadIdx.x >> 5;
  const unsigned lane = threadIdx.x & 31u;
  float* wbuf = &tile[wave][0];

  const unsigned waveGlobal = blockIdx.x * kWavesPerBlk + wave;
  const unsigned nWaves     = gridDim.x * kWavesPerBlk;

  float lsum = 0.0f;

  for (unsigned t = waveGlobal; t < tiles; t += nWaves) {
    const float* gbase = out + (size_t)t * (size_t)(kRowsPerTile * kC);
    unsigned rowsLeft = n - t * kRowsPerTile;
    if (rowsLeft > kRowsPerTile) rowsLeft = kRowsPerTile;

    // Issue the target load early: it is an ordinary LOADcnt global load and
    // completes under the cover of the async tile fill below.
    const unsigned row    = t * kRowsPerTile + lane;
    const bool     active = (row < n);
    int tv = 0;
    if (active) tv = tgt[row];

    // --- async fill: 27 x (32 lanes x 8 B), linear global addresses.
    // global float index of chunk cid is exactly 2*cid (216 = 27*8);
    // LDS float index = 2*cid + 4*(cid/27) (16 B row padding).
    if (rowsLeft == kRowsPerTile) {
      // full tile (the only case for n % 32 == 0): branch-free, fully unrolled
#pragma unroll
      for (int it = 0; it < kChunks; ++it) {
        unsigned cid = (unsigned)it * 32u + lane;
        unsigned r   = cid / (unsigned)kChunks;   // 0..31
        async_load_b64_to_lds(gbase + (size_t)cid * 2u,
                              wbuf + cid * 2u + r * 4u);
      }
    } else {
      // tail tile: per-chunk guard
#pragma unroll 1
      for (int it = 0; it < kChunks; ++it) {
        unsigned cid = (unsigned)it * 32u + lane;
        unsigned r   = cid / (unsigned)kChunks;
        if (r < rowsLeft) {
          async_load_b64_to_lds(gbase + (size_t)cid * 2u,
                                wbuf + cid * 2u + r * 4u);
        }
      }
    }
    wait_async0();

    // --- per-lane row scan from LDS (conflict-free b64 reads)
    if (active) {
      const float*  rp  = wbuf + lane * (unsigned)kLdsStride;   // 232 B stride, 8B aligned
      const float2* rp2 = reinterpret_cast<const float2*>(rp);

      float best = -__builtin_inff();
      int   bi   = 0;
#pragma unroll
      for (int j = 0; j < kChunks; ++j) {
        float2 v = rp2[j];
        if (v.x > best) { best = v.x; bi = 2 * j; }
        if (v.y > best) { best = v.y; bi = 2 * j + 1; }
      }

      const bool keep =
          (((kKeptMask >> (unsigned)bi) | (kKeptMask >> (unsigned)tv)) & 1ull) != 0ull;
      const float g = keep ? rp[tv] : adj;
      lsum += logf(g);
    }

    // buffer reuse guard: all LDS reads complete before next async fill
    wait_ds0();
  }

  // --- deterministic in-block tree reduction
  red[threadIdx.x] = lsum;
  __syncthreads();
#pragma unroll
  for (int s = 128; s > 0; s >>= 1) {
    if ((int)threadIdx.x < s) red[threadIdx.x] += red[threadIdx.x + s];
    __syncthreads();
  }
  if (threadIdx.x == 0) partials[blockIdx.x] = red[0];
}

// ---------------------------------------------------------------------------
// Kernel 2: final reduction of per-block partials in fp64, write scalar loss
// ---------------------------------------------------------------------------
__global__ void __launch_bounds__(256)
green_nll_final(const float* __restrict__ partials, int nblocks,
                float* __restrict__ out, double neg_inv_n) {
  __shared__ double dred[256];
  double s = 0.0;
  for (int i = (int)threadIdx.x; i < nblocks; i += 256) s += (double)partials[i];
  dred[threadIdx.x] = s;
  __syncthreads();
#pragma unroll
  for (int st = 128; st > 0; st >>= 1) {
    if ((int)threadIdx.x < st) dred[threadIdx.x] += dred[threadIdx.x + st];
    __syncthreads();
  }
  if (threadIdx.x == 0) out[0] = (float)(dred[0] * neg_inv_n);
}

// ---------------------------------------------------------------------------
extern "C" void kernel_launch(void* const* d_in, const int* in_sizes, int n_in,
                              void* d_out, int out_size, void* d_ws, size_t ws_size,
                              hipStream_t stream) {
  const float* out_probs = (const float*)d_in[0];   // [n, 54] fp32
  const int*   target    = (const int*)d_in[1];     // [n] int
  (void)n_in; (void)out_size; (void)ws_size;

  const unsigned n = (unsigned)in_sizes[1];
  const int      c = (int)(in_sizes[0] / (int)n);   // 54
  const float adj  = 1.0f - (float)(c - 1) * 0.001f;

  const unsigned tiles = (n + kRowsPerTile - 1) / kRowsPerTile;
  unsigned blocks = (tiles + kWavesPerBlk - 1) / kWavesPerBlk;
  if (blocks > 2048u) blocks = 2048u;
  if (blocks < 1u)    blocks = 1u;

  float* partials = (float*)d_ws;   // blocks floats <= 8 KB

  green_nll_main<<<dim3(blocks), dim3(256), 0, stream>>>(
      out_probs, target, partials, n, tiles, adj);
  green_nll_final<<<dim3(1), dim3(256), 0, stream>>>(
      partials, (int)blocks, (float*)d_out, -1.0 / (double)n);
}